// LengthRegulator_82188494176729
// MI455X (gfx1250) — compile-verified
//
#include <hip/hip_runtime.h>

#define B_ 16
#define C_ 256
#define T_ 512
#define LTILE 256
#define CCHUNK 64

#if defined(__has_builtin)
#if __has_builtin(__builtin_amdgcn_global_load_async_to_lds_b32)
#define LR_ASYNC_LDS 1
#endif
#if __has_builtin(__builtin_amdgcn_s_wait_asynccnt)
#define LR_HAVE_WAIT_ASYNC 1
#endif
#endif

typedef __attribute__((address_space(1))) int* gint_ptr;
typedef __attribute__((address_space(3))) int* lint_ptr;

// ---------------------------------------------------------------------------
// Kernel 1: per-batch inclusive scan of durations (tiny: 16 x 512 int32).
// One lane per batch. Also emits mel_len (as float) into its output slot.
// ---------------------------------------------------------------------------
__global__ void lr_cumsum(const int* __restrict__ dur,
                          int* __restrict__ cum,
                          float* __restrict__ mel_out) {
  const int b = threadIdx.x;
  if (b >= B_) return;
  const int* __restrict__ d = dur + b * T_;
  int* __restrict__ c = cum + b * T_;
  int s = 0;
  for (int t = 0; t < T_; ++t) {
    int v = d[t];
    v = v < 0 ? 0 : v;          // jnp.maximum(duration, 0)
    s += v;
    c[t] = s;                   // inclusive cumsum (max 7*512 = 3584, fits int)
  }
  mel_out[b] = (float)s;        // mel_len as float (output buffer is float)
}

// ---------------------------------------------------------------------------
// Kernel 2: gather/expand. Block = 256 threads over frames l; grid.y = batch,
// grid.z = channel chunk. Stage cum[b][0..511] (2KB) into LDS via gfx1250
// async global->LDS loads when available, binary-search LDS (searchsorted
// side='right'), then stream CCHUNK channels with coalesced stores.
// ---------------------------------------------------------------------------
__global__ void lr_gather(const float* __restrict__ x,
                          const int* __restrict__ cum,
                          float* __restrict__ out,
                          int L) {
  __shared__ int s_cum[T_];

  const int b  = blockIdx.y;
  const int c0 = blockIdx.z * CCHUNK;
  const int l  = blockIdx.x * LTILE + threadIdx.x;

  const int* __restrict__ gcum = cum + b * T_;

#if defined(LR_ASYNC_LDS)
  // gfx1250 async path: GLOBAL_LOAD_ASYNC_TO_LDS_B32 (tracked by ASYNCcnt)
  for (int i = threadIdx.x; i < T_; i += LTILE) {
    __builtin_amdgcn_global_load_async_to_lds_b32(
        (gint_ptr)(gcum + i),
        (lint_ptr)(&s_cum[i]),
        0, 0);
  }
#if defined(LR_HAVE_WAIT_ASYNC)
  __builtin_amdgcn_s_wait_asynccnt(0);
#else
  asm volatile("s_wait_asynccnt 0" ::: "memory");
#endif
#else
  for (int i = threadIdx.x; i < T_; i += LTILE) {
    s_cum[i] = gcum[i];
  }
#endif
  __syncthreads();

  if (l >= L) return;

  // searchsorted(cum, l, side='right'): first index with cum[idx] > l.
  int lo = 0, hi = T_;
  while (lo < hi) {
    const int mid = (lo + hi) >> 1;
    if (s_cum[mid] <= l) lo = mid + 1; else hi = mid;
  }
  const int idx = (lo < T_) ? lo : (T_ - 1);   // clip to T-1
  const int mel = s_cum[T_ - 1];
  const bool valid = (l < mel);

  const float* __restrict__ xb =
      x + (size_t)b * (size_t)C_ * T_ + (size_t)c0 * T_ + idx;
  float* __restrict__ ob =
      out + (size_t)b * (size_t)C_ * L + (size_t)c0 * L + l;

  #pragma unroll 8
  for (int c = 0; c < CCHUNK; ++c) {
    // speculative read is safe: idx clamped in-bounds even for padded frames
    const float v = valid ? xb[(size_t)c * T_] : 0.0f;
    ob[(size_t)c * L] = v;
  }
}

// ---------------------------------------------------------------------------
// Host launcher
// ---------------------------------------------------------------------------
extern "C" void kernel_launch(void* const* d_in, const int* in_sizes, int n_in,
                              void* d_out, int out_size, void* d_ws, size_t ws_size,
                              hipStream_t stream) {
  const float* x   = (const float*)d_in[0];   // [B, C, T] float32
  const int*   dur = (const int*)d_in[1];     // [B, T] int
  float* out = (float*)d_out;                 // [B, C, L] floats ++ [B] mel_len
  int*   cum = (int*)d_ws;                    // B*T int32 scratch (32 KB)

  const int L = (out_size - B_) / (B_ * C_);
  float* mel_out = out + (size_t)B_ * (size_t)C_ * (size_t)L;

  lr_cumsum<<<dim3(1), dim3(32), 0, stream>>>(dur, cum, mel_out);

  dim3 grid((L + LTILE - 1) / LTILE, B_, C_ / CCHUNK);
  lr_gather<<<grid, dim3(LTILE), 0, stream>>>(x, cum, out, L);
}